// SceneGraph2D_24120536334640
// MI455X (gfx1250) — compile-verified
//
#include <hip/hip_runtime.h>

typedef __attribute__((ext_vector_type(2))) float v2f;
typedef __attribute__((ext_vector_type(8))) float v8f;

#define B_    8
#define C_    256
#define H_    16
#define W_    24
#define HW_   384
#define N_    16
#define P_    7
#define PP_   49
#define C32_  384
#define D1_   256
#define D2_   256
#define IMGW_ 384.0f
#define IMGH_ 256.0f
#define SCALE_ (1.0f/16.0f)

// ---------------------------------------------------------------------------
// helpers: PrRoI hat kernel integration and intersection maps
// ---------------------------------------------------------------------------
__device__ __forceinline__ float hatG(float t) {
  float tl = fminf(fmaxf(t, -1.f), 0.f);
  float tr = fminf(fmaxf(t,  0.f), 1.f);
  return 0.5f*(tl+1.f)*(tl+1.f) + 0.5f - 0.5f*(1.f-tr)*(1.f-tr);
}

// out[q*nb + j] = G(e_{q+1} - j) - G(e_q - j), e_i = e1 + (e2-e1)*i/P
__device__ void bin_weights(float e1, float e2, int nb, float* out) {
  for (int q = 0; q < P_; ++q) {
    float ea = e1 + (e2 - e1) * ((float)q       / (float)P_);
    float eb = e1 + (e2 - e1) * ((float)(q + 1) / (float)P_);
    for (int j = 0; j < nb; ++j)
      out[q*nb + j] = hatG(eb - (float)j) - hatG(ea - (float)j);
  }
}

// intersection_map(inp, anchor) -> out[49]
__device__ void imap_fill(float ix1, float iy1, float ix2, float iy2,
                          float ax1, float ay1, float ax2, float ay2,
                          float* out) {
  float cw = fmaxf((ax2-ax1)/(float)P_, 1e-9f);
  float ch = fmaxf((ay2-ay1)/(float)P_, 1e-9f);
  float ox[P_], oy[P_];
  for (int q = 0; q < P_; ++q) {
    float e0 = ax1 + (ax2-ax1)*((float)q    /(float)P_);
    float e1 = ax1 + (ax2-ax1)*((float)(q+1)/(float)P_);
    ox[q] = fmaxf(fminf(e1, ix2) - fmaxf(e0, ix1), 0.f);
  }
  for (int p = 0; p < P_; ++p) {
    float e0 = ay1 + (ay2-ay1)*((float)p    /(float)P_);
    float e1 = ay1 + (ay2-ay1)*((float)(p+1)/(float)P_);
    oy[p] = fmaxf(fminf(e1, iy2) - fmaxf(e0, iy1), 0.f);
  }
  for (int p = 0; p < P_; ++p)
    for (int q = 0; q < P_; ++q)
      out[p*P_+q] = (oy[p]/ch) * (ox[q]/cw);
}

// ---------------------------------------------------------------------------
// per-box precompute kernels
// ---------------------------------------------------------------------------
__global__ void boxprep_kernel(const float* __restrict__ boxes,
                               float* __restrict__ wy, float* __restrict__ wx,
                               float* __restrict__ inva, float* __restrict__ imap) {
  int g = threadIdx.x;
  if (g >= B_*N_) return;
  float x1 = boxes[g*4+0], y1 = boxes[g*4+1], x2 = boxes[g*4+2], y2 = boxes[g*4+3];
  float sx1 = x1*SCALE_, sy1 = y1*SCALE_, sx2 = x2*SCALE_, sy2 = y2*SCALE_;
  bin_weights(sy1, sy2, H_, wy + (long)g*P_*H_);
  bin_weights(sx1, sx2, W_, wx + (long)g*P_*W_);
  float area = (sx2-sx1)*(sy2-sy1)/(float)PP_;
  inva[g] = (area > 0.f) ? 1.f/fmaxf(area, 1e-9f) : 0.f;
  imap_fill(x1, y1, x2, y2, 0.f, 0.f, IMGW_, IMGH_, imap + (long)g*PP_);
}

__global__ void imgprep_kernel(float* __restrict__ wy, float* __restrict__ wx,
                               float* __restrict__ inva) {
  int t = threadIdx.x;
  if (t >= B_) return;
  bin_weights(0.f, IMGH_*SCALE_, H_, wy + (long)t*P_*H_);
  bin_weights(0.f, IMGW_*SCALE_, W_, wx + (long)t*P_*W_);
  float area = (IMGW_*SCALE_)*(IMGH_*SCALE_)/(float)PP_;
  inva[t] = 1.f/fmaxf(area, 1e-9f);
}

__global__ void unionprep_kernel(const float* __restrict__ boxes,
                                 float* __restrict__ wy, float* __restrict__ wx,
                                 float* __restrict__ inva,
                                 float* __restrict__ simap, float* __restrict__ oimap) {
  int g = blockIdx.x*blockDim.x + threadIdx.x;
  if (g >= B_*N_*N_) return;
  int b = g >> 8, pair = g & 255;
  int s = pair >> 4, o = pair & 15;
  const float* sb = boxes + (long)(b*N_+s)*4;
  const float* ob = boxes + (long)(b*N_+o)*4;
  float ux1 = fminf(sb[0], ob[0]);
  float uy1 = fminf(sb[1], ob[1]);
  float ux2 = fmaxf(sb[2], ob[2]);
  float uy2 = fmaxf(sb[3], ob[3]);
  bin_weights(uy1*SCALE_, uy2*SCALE_, H_, wy + (long)g*P_*H_);
  bin_weights(ux1*SCALE_, ux2*SCALE_, W_, wx + (long)g*P_*W_);
  float area = (ux2-ux1)*SCALE_*(uy2-uy1)*SCALE_/(float)PP_;
  inva[g] = (area > 0.f) ? 1.f/fmaxf(area, 1e-9f) : 0.f;
  imap_fill(sb[0], sb[1], sb[2], sb[3], ux1, uy1, ux2, uy2, simap + (long)g*PP_);
  imap_fill(ob[0], ob[1], ob[2], ob[3], ux1, uy1, ux2, uy2, oimap + (long)g*PP_);
}

// ---------------------------------------------------------------------------
// generic separable PrRoI pooling: out[box, c, p, q]
// ---------------------------------------------------------------------------
__global__ __launch_bounds__(256)
void pool_kernel(const float* __restrict__ feat, long img_stride, int C, int boxes_per_img,
                 const float* __restrict__ wy, const float* __restrict__ wx,
                 const float* __restrict__ inva,
                 float* __restrict__ out, long box_stride, long ch_off) {
  int box = blockIdx.x;
  __shared__ float s_wy[P_*H_];
  __shared__ float s_wx[P_*W_];
  for (int i = threadIdx.x; i < P_*H_; i += blockDim.x) s_wy[i] = wy[(long)box*P_*H_ + i];
  for (int i = threadIdx.x; i < P_*W_; i += blockDim.x) s_wx[i] = wx[(long)box*P_*W_ + i];
  __syncthreads();
  float inv = inva[box];
  int img = box / boxes_per_img;
  const float* Fb = feat + (long)img*img_stride;
  float* Ob = out + (long)box*box_stride + ch_off*PP_;
  for (int c = threadIdx.x; c < C; c += blockDim.x) {
    const float* F = Fb + (long)c*HW_;
    float* O = Ob + (long)c*PP_;
    for (int p = 0; p < P_; ++p) {
      float tw[W_];
      #pragma unroll
      for (int w = 0; w < W_; ++w) tw[w] = 0.f;
      for (int h = 0; h < H_; ++h) {
        float cy = s_wy[p*H_ + h];
        #pragma unroll
        for (int w = 0; w < W_; ++w) tw[w] += cy * F[h*W_ + w];
      }
      for (int q = 0; q < P_; ++q) {
        float acc = 0.f;
        #pragma unroll
        for (int w = 0; w < W_; ++w) acc += s_wx[q*W_ + w] * tw[w];
        O[p*P_+q] = acc * inv;
      }
    }
  }
}

// ---------------------------------------------------------------------------
// generic WMMA f32 GEMM: D[m,n] = sum_k A[m,k] * f(B[k,n]) + bias[m]
//   Requirements (all call sites obey): K % 8 == 0, M % 16 == 0, lda even,
//   A/B bases 8-byte aligned (workspace carve aligns to 16 B).
//   BK1 != 0 asserts ldbk == 1 (FC layout): B k-pairs load as packed v2f.
//   Out-of-range tile rows/cols are CLAMPED, not masked: an invalid row m /
//   col n of the 16x16 WMMA tile only feeds D elements that the store loop
//   discards, so clamped (legal-address) garbage is harmless. This keeps the
//   inner loop free of exec-mask predication.
// ---------------------------------------------------------------------------
template <int RELUB, int BK1>
__global__ __launch_bounds__(256)
void gemm_wmma_f32(const float* __restrict__ A, long lda, long strideA,
                   const float* __restrict__ Bm, long ldbk, long ldbn, long strideB,
                   const float* __restrict__ bias,
                   float* __restrict__ Cm, long ldcm, long ldcn, long strideC,
                   int M, int N, int K) {
  int wave = threadIdx.x >> 5;
  int lane = threadIdx.x & 31;
  int Mt = (M + 15) >> 4, Nt = (N + 15) >> 4;
  int tile = blockIdx.x * (blockDim.x >> 5) + wave;
  if (tile >= Mt*Nt) return;                 // wave-uniform exit (EXEC stays full)
  int tm = (tile / Nt) << 4;
  int tn = (tile % Nt) << 4;
  long zb = blockIdx.z;
  const float* Ab = A  + zb*strideA;
  const float* Bb = Bm + zb*strideB;
  float*       Cb = Cm + zb*strideC;
  int r  = lane & 15;
  int hi = lane >> 4;
  int am = min(tm + r, M - 1);               // clamp (see header comment)
  int bn = min(tn + r, N - 1);
  // A lanes: row m = r, k pair = {ka, ka+1}, ka = k0 + 2*hi (ISA 16x4 layout)
  // B lanes: col n = r, same k pair       (ISA 4x16 layout)
  const float* pa = Ab + (long)am*lda + (hi << 1);
  const float* pb = Bb + (long)bn*ldbn + (long)(hi << 1)*ldbk;
  const long bstep = 4*ldbk;
  v8f acc = {};
  for (int k0 = 0; k0 < K; k0 += 8) {
    v2f a0 = *(const v2f*)pa;
    v2f a1 = *(const v2f*)(pa + 4);
    v2f b0, b1;
    if (BK1) {
      b0 = *(const v2f*)pb;
      b1 = *(const v2f*)(pb + 4);
    } else {
      b0.x = pb[0];     b0.y = pb[ldbk];
      b1.x = pb[bstep]; b1.y = pb[bstep + ldbk];
    }
    if (RELUB) {
      b0.x = fmaxf(b0.x, 0.f); b0.y = fmaxf(b0.y, 0.f);
      b1.x = fmaxf(b1.x, 0.f); b1.y = fmaxf(b1.y, 0.f);
    }
    acc = __builtin_amdgcn_wmma_f32_16x16x4_f32(false, a0, false, b0, (short)0, acc,
                                                false, false);
    acc = __builtin_amdgcn_wmma_f32_16x16x4_f32(false, a1, false, b1, (short)0, acc,
                                                false, false);
    pa += 8;
    pb += 2*bstep;
  }
  #pragma unroll
  for (int g = 0; g < 8; ++g) {
    int m = tm + g + (hi << 3);              // D: VGPR g -> row g (lanes 0-15), g+8 (16-31)
    int n = tn + r;
    if (m < M && n < N) {
      float v = acc[g];
      if (bias) v += bias[m];
      Cb[(long)m*ldcm + (long)n*ldcn] = v;
    }
  }
}

// ---------------------------------------------------------------------------
// concat builders
// ---------------------------------------------------------------------------
__global__ void objin_kernel(const float* __restrict__ obj_pool,
                             const float* __restrict__ ctx_pool,
                             const float* __restrict__ boximap,
                             float* __restrict__ obj_in) {
  long idx = (long)blockIdx.x*blockDim.x + threadIdx.x;
  long total = (long)B_*N_*512*PP_;
  if (idx >= total) return;
  int g  = (int)(idx / (512*PP_));           // box 0..127
  int j  = (int)(idx % (512*PP_));
  int ch = j / PP_, pq = j % PP_;
  int b  = g >> 4;
  float v;
  if (ch < 256) v = obj_pool[(long)g*(C_*PP_) + j];
  else {
    v = ctx_pool[(long)b*(C_*PP_) + (ch-256)*PP_ + pq];   // x (0..127) / y (128..255)
    if (ch >= 384) v *= boximap[(long)g*PP_ + pq];        // y * box_imap
  }
  obj_in[idx] = v;
}

__global__ void relcat_kernel(const float* __restrict__ obj,     // [N,256,49] this image
                              const float* __restrict__ simap,   // [256,49] this image
                              const float* __restrict__ oimap,
                              float* __restrict__ relin) {       // [256][896][49]
  int pair = blockIdx.x;
  int s = pair >> 4, o = pair & 15;
  float* R = relin + (long)pair*(896*PP_);
  const float* Fs = obj + (long)s*(D1_*PP_);
  const float* Fo = obj + (long)o*(D1_*PP_);
  for (int j = threadIdx.x; j < 512*PP_; j += blockDim.x)
    R[j] = (j < D1_*PP_) ? Fs[j] : Fo[j - D1_*PP_];
  const float* si = simap + (long)pair*PP_;
  const float* oi = oimap + (long)pair*PP_;
  for (int j = threadIdx.x; j < 128*PP_; j += blockDim.x) {
    int pq = j % PP_;
    R[640*PP_ + j] *= si[pq];   // ry * sub_imap
    R[768*PP_ + j] *= oi[pq];   // rz * obj_imap
  }
}

// ---------------------------------------------------------------------------
// row-wise L2 normalize (dim = 256)
// ---------------------------------------------------------------------------
__global__ __launch_bounds__(256)
void l2norm_kernel(const float* __restrict__ in, float* __restrict__ out) {
  int row = blockIdx.x;
  __shared__ float red[256];
  float v = in[(long)row*256 + threadIdx.x];
  red[threadIdx.x] = v*v;
  __syncthreads();
  for (int s = 128; s > 0; s >>= 1) {
    if (threadIdx.x < s) red[threadIdx.x] += red[threadIdx.x + s];
    __syncthreads();
  }
  float nrm = sqrtf(red[0]);
  out[(long)row*256 + threadIdx.x] = v / nrm;
}

// ---------------------------------------------------------------------------
extern "C" void kernel_launch(void* const* d_in, const int* in_sizes, int n_in,
                              void* d_out, int out_size, void* d_ws, size_t ws_size,
                              hipStream_t stream) {
  (void)in_sizes; (void)n_in; (void)out_size; (void)ws_size;
  const float* input   = (const float*)d_in[0];
  const float* objects = (const float*)d_in[1];
  const float* W_ctx   = (const float*)d_in[3];
  const float* b_ctx   = (const float*)d_in[4];
  const float* W_rel   = (const float*)d_in[5];
  const float* b_rel   = (const float*)d_in[6];
  const float* W_of    = (const float*)d_in[7];
  const float* b_of    = (const float*)d_in[8];
  const float* W_rf    = (const float*)d_in[9];
  const float* b_rf    = (const float*)d_in[10];
  const float* W_ofc   = (const float*)d_in[11];
  const float* b_ofc   = (const float*)d_in[12];
  const float* W_rfc   = (const float*)d_in[13];
  const float* b_rfc   = (const float*)d_in[14];
  float* out = (float*)d_out;

  // workspace carve-out (~97 MB of floats); every sub-array 16-byte aligned
  float* ws = (float*)d_ws;
  size_t off = 0;
  auto carve = [&](size_t n) { float* p = ws + off; off += (n + 3) & ~(size_t)3; return p; };
  float* ctx_feat = carve((size_t)B_*C_*HW_);          // [B,256,384]
  float* rel_feat = carve((size_t)B_*C32_*HW_);        // [B,384,384]
  float* wy_obj   = carve((size_t)B_*N_*P_*H_);
  float* wx_obj   = carve((size_t)B_*N_*P_*W_);
  float* inva_obj = carve((size_t)B_*N_);
  float* boximap  = carve((size_t)B_*N_*PP_);
  float* wy_img   = carve((size_t)B_*P_*H_);
  float* wx_img   = carve((size_t)B_*P_*W_);
  float* inva_img = carve((size_t)B_);
  float* ctx_pool = carve((size_t)B_*C_*PP_);
  float* obj_pool = carve((size_t)B_*N_*C_*PP_);       // [B,N,256,49]
  float* obj_in   = carve((size_t)B_*N_*512*PP_);      // [B,N,512,49]
  float* obj_buf  = carve((size_t)B_*N_*D1_*PP_);      // [B,N,256,49]
  float* obj_raw  = carve((size_t)B_*N_*D1_);          // [B,N,256]
  float* wy_un    = carve((size_t)B_*256*P_*H_);
  float* wx_un    = carve((size_t)B_*256*P_*W_);
  float* inva_un  = carve((size_t)B_*256);
  float* simap    = carve((size_t)B_*256*PP_);
  float* oimap    = carve((size_t)B_*256*PP_);
  float* relin    = carve((size_t)256*896*PP_);        // per-image [256,896,49]
  float* relconv  = carve((size_t)256*D2_*PP_);        // per-image [256,256,49]
  float* rel_raw  = carve((size_t)B_*256*D2_);         // [B,256,256]

  auto gemm = [&](const float* A, long lda, long sA,
                  const float* Bm, long ldbk, long ldbn, long sB,
                  const float* bias, float* Cm, long ldcm, long ldcn, long sC,
                  int M, int N, int K, int reluB, int batch) {
    int Mt = (M + 15) / 16, Nt = (N + 15) / 16;
    int tiles = Mt * Nt;
    dim3 grid((tiles + 7) / 8, 1, batch);
    if (reluB)   // FC calls: ldbk == 1 -> packed B loads
      gemm_wmma_f32<1, 1><<<grid, 256, 0, stream>>>(A, lda, sA, Bm, ldbk, ldbn, sB,
                                                    bias, Cm, ldcm, ldcn, sC, M, N, K);
    else
      gemm_wmma_f32<0, 0><<<grid, 256, 0, stream>>>(A, lda, sA, Bm, ldbk, ldbn, sB,
                                                    bias, Cm, ldcm, ldcn, sC, M, N, K);
  };

  // 1) per-box weights / intersection maps
  boxprep_kernel<<<1, 128, 0, stream>>>(objects, wy_obj, wx_obj, inva_obj, boximap);
  imgprep_kernel<<<1, 32, 0, stream>>>(wy_img, wx_img, inva_img);
  unionprep_kernel<<<8, 256, 0, stream>>>(objects, wy_un, wx_un, inva_un, simap, oimap);

  // 2) 1x1 convs: ctx_feat = W_ctx @ feat + b ; rel_feat = W_rel @ feat + b
  gemm(W_ctx, 256, 0, input, HW_, 1, (long)C_*HW_, b_ctx,
       ctx_feat, HW_, 1, (long)C_*HW_, C_, HW_, C_, 0, B_);
  gemm(W_rel, 256, 0, input, HW_, 1, (long)C_*HW_, b_rel,
       rel_feat, HW_, 1, (long)C32_*HW_, C32_, HW_, C_, 0, B_);

  // 3) pooling: ctx pooled with full-image box (once per image), obj boxes on raw feat
  pool_kernel<<<B_, 256, 0, stream>>>(ctx_feat, (long)C_*HW_, C_, 1,
                                      wy_img, wx_img, inva_img,
                                      ctx_pool, (long)C_*PP_, 0);
  pool_kernel<<<B_*N_, 256, 0, stream>>>(input, (long)C_*HW_, C_, N_,
                                         wy_obj, wx_obj, inva_obj,
                                         obj_pool, (long)C_*PP_, 0);

  // 4) obj conv: concat([obj_pool, x, y*box_imap]) -> W_objfuse
  {
    long total = (long)B_*N_*512*PP_;
    objin_kernel<<<(int)((total + 255) / 256), 256, 0, stream>>>(obj_pool, ctx_pool,
                                                                 boximap, obj_in);
  }
  gemm(W_of, 512, 0, obj_in, PP_, 1, (long)512*PP_, b_of,
       obj_buf, PP_, 1, (long)D1_*PP_, D1_, PP_, 512, 0, B_*N_);

  // 5) obj fc: relu(obj flat [N,12544]) @ W_objfc^T + b, per image; then l2norm
  gemm(W_ofc, 12544, 0, obj_buf, 1, 12544, (long)N_*D1_*PP_, b_ofc,
       obj_raw, 1, D1_, (long)N_*D1_, D1_, N_, 12544, 1, B_);
  l2norm_kernel<<<B_*N_, 256, 0, stream>>>(obj_raw, out);

  // 6) relation path, per image (relin/relconv scratch reused)
  for (int b = 0; b < B_; ++b) {
    pool_kernel<<<256, 256, 0, stream>>>(rel_feat + (long)b*C32_*HW_, 0, C32_, 256,
                                         wy_un + (long)b*256*P_*H_,
                                         wx_un + (long)b*256*P_*W_,
                                         inva_un + (long)b*256,
                                         relin, (long)896*PP_, 512);
    relcat_kernel<<<256, 256, 0, stream>>>(obj_buf + (long)b*N_*D1_*PP_,
                                           simap + (long)b*256*PP_,
                                           oimap + (long)b*256*PP_, relin);
    gemm(W_rf, 896, 0, relin, PP_, 1, (long)896*PP_, b_rf,
         relconv, PP_, 1, (long)D2_*PP_, D2_, PP_, 896, 0, 256);
    gemm(W_rfc, 12544, 0, relconv, 1, 12544, 0, b_rfc,
         rel_raw + (long)b*256*D2_, 1, D2_, 0, D2_, 256, 12544, 1, 1);
  }
  l2norm_kernel<<<B_*256, 256, 0, stream>>>(rel_raw, out + (long)B_*N_*D1_);
}